// CazzyLoss_35862976921713
// MI455X (gfx1250) — compile-verified
//
#include <hip/hip_runtime.h>
#include <hip/hip_bf16.h>

#define DVOCAB 1400

typedef __attribute__((ext_vector_type(2))) float v2f;
typedef __attribute__((ext_vector_type(8))) float v8f;

__device__ inline float waveReduceSum(float v) {
    #pragma unroll
    for (int o = 16; o > 0; o >>= 1) v += __shfl_xor(v, o, 32);
    return v;
}
__device__ inline float waveReduceMax(float v) {
    #pragma unroll
    for (int o = 16; o > 0; o >>= 1) v = fmaxf(v, __shfl_xor(v, o, 32));
    return v;
}
__device__ inline int waveReduceSumI(int v) {
    #pragma unroll
    for (int o = 16; o > 0; o >>= 1) v += __shfl_xor(v, o, 32);
    return v;
}

// ---------------------------------------------------------------------------
// Workspace layout (floats):
//   acc[0] disease nll sum     acc[1] disease valid count
//   acc[2] time term sum       acc[3] risk nll sum
//   acc[4] risk valid count    acc[5] uncertainty sum
//   acc[6] concordance sum     word 8: (unsigned) pair count
//   words 16.. : m[n] survival means
// ---------------------------------------------------------------------------

__global__ void zero_acc_kernel(float* acc) {
    if (threadIdx.x < 16) acc[threadIdx.x] = 0.0f;
}

// One wave (32 lanes) per row of 1400 logits. Single pass: row held in VGPRs.
__global__ void disease_ce_kernel(const float* __restrict__ logits,
                                  const int* __restrict__ tgt,
                                  float* __restrict__ acc, int nRows) {
    const int lane = threadIdx.x & 31;
    const int wave = threadIdx.x >> 5;
    const int row  = blockIdx.x * (blockDim.x >> 5) + wave;
    if (row >= nRows) return;

    const float* rp = logits + (size_t)row * DVOCAB;

    // 10 x float4 per lane covers idx 0..1279 (16B-aligned: 1400*4 % 16 == 0)
    float4 v4[10];
    #pragma unroll
    for (int it = 0; it < 10; ++it)
        v4[it] = *(const float4*)(rp + it * 128 + lane * 4);
    // scalar tail 1280..1399
    float vt[4];
    #pragma unroll
    for (int k = 0; k < 4; ++k) {
        int idx = 1280 + k * 32 + lane;
        vt[k] = (idx < DVOCAB) ? rp[idx] : -3.402823466e38f;
    }

    float mx = -3.402823466e38f;
    #pragma unroll
    for (int it = 0; it < 10; ++it) {
        mx = fmaxf(mx, fmaxf(fmaxf(v4[it].x, v4[it].y), fmaxf(v4[it].z, v4[it].w)));
    }
    #pragma unroll
    for (int k = 0; k < 4; ++k) mx = fmaxf(mx, vt[k]);
    mx = waveReduceMax(mx);

    const int t = tgt[row];
    const bool valid = (t >= 0) && (t < DVOCAB);   // ignore_index=-100 -> invalid

    float se = 0.0f;   // sum exp(x - mx)
    float tl = 0.0f;   // logit at target (exactly one lane/slot matches)
    #pragma unroll
    for (int it = 0; it < 10; ++it) {
        int base = it * 128 + lane * 4;
        float x0 = v4[it].x, x1 = v4[it].y, x2 = v4[it].z, x3 = v4[it].w;
        se += expf(x0 - mx) + expf(x1 - mx) + expf(x2 - mx) + expf(x3 - mx);
        if (base + 0 == t) tl = x0;
        if (base + 1 == t) tl = x1;
        if (base + 2 == t) tl = x2;
        if (base + 3 == t) tl = x3;
    }
    #pragma unroll
    for (int k = 0; k < 4; ++k) {
        int idx = 1280 + k * 32 + lane;
        if (idx < DVOCAB) {
            se += expf(vt[k] - mx);
            if (idx == t) tl = vt[k];
        }
    }
    se = waveReduceSum(se);
    tl = waveReduceSum(tl);

    if (lane == 0 && valid) {
        float nll = -(tl - mx - logf(se));
        atomicAdd(&acc[0], nll);
        atomicAdd(&acc[1], 1.0f);
    }
}

// time term + uncertainty sum, one thread per element
__global__ void time_unc_kernel(const float* __restrict__ tte,
                                const float* __restrict__ ttg,
                                const float* __restrict__ unc,
                                float* __restrict__ acc, int N) {
    int i = blockIdx.x * blockDim.x + threadIdx.x;
    float term = 0.0f, u = 0.0f;
    if (i < N) {
        float r = 1.0f / (tte[i] + 1e-6f);
        term = logf(r + 1e-6f) - r * ttg[i];
        u = unc[i];
    }
    term = waveReduceSum(term);
    u = waveReduceSum(u);
    if ((threadIdx.x & 31) == 0) {
        atomicAdd(&acc[2], term);
        atomicAdd(&acc[5], u);
    }
}

// risk CE: 5 classes, one thread per row
__global__ void risk_ce_kernel(const float* __restrict__ logits,
                               const int* __restrict__ tgt,
                               float* __restrict__ acc, int nRows) {
    int i = blockIdx.x * blockDim.x + threadIdx.x;
    float nll = 0.0f, validf = 0.0f;
    if (i < nRows) {
        const float* rp = logits + (size_t)i * 5;
        float x0 = rp[0], x1 = rp[1], x2 = rp[2], x3 = rp[3], x4 = rp[4];
        float mx = fmaxf(fmaxf(fmaxf(x0, x1), fmaxf(x2, x3)), x4);
        float se = expf(x0 - mx) + expf(x1 - mx) + expf(x2 - mx) +
                   expf(x3 - mx) + expf(x4 - mx);
        int t = tgt[i];
        if (t >= 0 && t < 5) {
            float tv = x0;
            if (t == 1) tv = x1;
            if (t == 2) tv = x2;
            if (t == 3) tv = x3;
            if (t == 4) tv = x4;
            nll = -(tv - mx - logf(se));
            validf = 1.0f;
        }
    }
    nll = waveReduceSum(nll);
    validf = waveReduceSum(validf);
    if ((threadIdx.x & 31) == 0) {
        atomicAdd(&acc[3], nll);
        atomicAdd(&acc[4], validf);
    }
}

// Survival row means via V_WMMA_F32_16X16X4_F32 with B = ones (exact fp32
// row-sum of a 16-row tile; each K-step consumes 4 columns).
// One wave per 16-row tile; EXEC all-ones through the WMMA loop.
__global__ void surv_mean_wmma_kernel(const float* __restrict__ S,
                                      float* __restrict__ m, int n, int T) {
    const int lane = threadIdx.x;            // blockDim.x == 32
    const int tile = blockIdx.x;
    const int rowInTile = lane & 15;
    const int kbase = (lane >> 4) * 2;       // lanes 0-15: K+0..1, lanes 16-31: K+2..3
    int row = tile * 16 + rowInTile;
    if (row >= n) row = n - 1;               // clamp (uniform-safe, no EXEC change)
    const float* rp = S + (size_t)row * T;

    v8f c = {};
    v2f b; b[0] = 1.0f; b[1] = 1.0f;
    const int T4 = T & ~3;
    for (int k = 0; k < T4; k += 4) {
        float2 ld = *(const float2*)(rp + k + kbase);
        v2f a; a[0] = ld.x; a[1] = ld.y;
        // (neg_a, A, neg_b, B, c_mod, C, reuse_a, reuse_b)
        c = __builtin_amdgcn_wmma_f32_16x16x4_f32(
                false, a, false, b, (short)0, c, false, false);
    }

    // D layout: VGPR v holds M=v (lanes 0-15) / M=v+8 (lanes 16-31), N=lane%16.
    // Every N-column is identical (B all ones) -> lanes 0 and 16 write 8 rows each.
    if ((lane & 15) == 0) {
        const int rbase = tile * 16 + (lane >> 4) * 8;
        const float inv = 1.0f / (float)T;
        #pragma unroll
        for (int v = 0; v < 8; ++v) {
            int r = rbase + v;
            if (r < n) {
                float s = c[v];
                for (int k = T4; k < T; ++k) s += S[(size_t)r * T + k];
                m[r] = s * inv;
            }
        }
    }
}

// Tiled O(n^2) concordance: block (iTile,jTile); j tile staged in LDS.
__global__ void concordance_kernel(const float* __restrict__ m,
                                   const float* __restrict__ t,
                                   const int* __restrict__ e,
                                   float* __restrict__ acc,
                                   unsigned* __restrict__ paircnt, int n) {
    const int iTile = blockIdx.x, jTile = blockIdx.y;
    if (jTile < iTile) return;               // whole block has j < i

    __shared__ float tj[256];
    __shared__ float mj[256];
    const int tid = threadIdx.x;
    const int j0 = jTile * 256;
    {
        int jg = j0 + tid;
        tj[tid] = (jg < n) ? t[jg] : -3.0e38f;   // padding: ti < tj always false
        mj[tid] = (jg < n) ? m[jg] : 0.0f;
    }
    __syncthreads();

    const int i = iTile * 256 + tid;
    float conc = 0.0f;
    int cnt = 0;
    if (i < n && e[i] == 1) {
        const float ti = t[i];
        const float mi = m[i];
        for (int jj = 0; jj < 256; ++jj) {
            int j = j0 + jj;
            if (j > i && j < n && ti < tj[jj]) {
                cnt++;
                float mjv = mj[jj];
                conc += (mi < mjv) ? 1.0f : ((mi == mjv) ? 0.5f : 0.0f);
            }
        }
    }
    conc = waveReduceSum(conc);
    cnt  = waveReduceSumI(cnt);
    if ((tid & 31) == 0 && (cnt > 0 || conc != 0.0f)) {
        atomicAdd(&acc[6], conc);
        atomicAdd(paircnt, (unsigned)cnt);
    }
}

__global__ void finalize_kernel(const float* __restrict__ acc,
                                const unsigned* __restrict__ paircnt,
                                float* __restrict__ out, int Nbs) {
    if (threadIdx.x == 0 && blockIdx.x == 0) {
        float disease = acc[0] / fmaxf(acc[1], 1.0f);
        float timel   = -acc[2] / (float)Nbs;
        float risk    = acc[3] / fmaxf(acc[4], 1.0f);
        unsigned p    = *paircnt;
        float surv    = (p > 0) ? (1.0f - acc[6] / (float)p) : 0.0f;
        float unc     = (acc[5] / (float)Nbs) * 0.01f;
        out[0] = disease;
        out[1] = timel;
        out[2] = risk;
        out[3] = surv;
        out[4] = unc;
        out[5] = disease + timel + risk + surv + unc;
    }
}

extern "C" void kernel_launch(void* const* d_in, const int* in_sizes, int n_in,
                              void* d_out, int out_size, void* d_ws, size_t ws_size,
                              hipStream_t stream) {
    const float* disease_logits  = (const float*)d_in[0];
    const int*   disease_targets = (const int*)  d_in[1];
    const float* tte             = (const float*)d_in[2];
    const float* ttg             = (const float*)d_in[3];
    const float* risk_logits     = (const float*)d_in[4];
    const int*   risk_targets    = (const int*)  d_in[5];
    const float* surv_curves     = (const float*)d_in[6];
    const float* surv_times      = (const float*)d_in[7];
    const int*   events          = (const int*)  d_in[8];
    const float* uncertainty     = (const float*)d_in[9];
    float* out = (float*)d_out;

    const int nRowsD = in_sizes[0] / DVOCAB;   // 16384
    const int Nbs    = in_sizes[2];            // 16384
    const int nRowsR = in_sizes[4] / 5;        // 16384
    const int n      = in_sizes[7];            // 4096
    const int T      = in_sizes[6] / n;        // 120

    float*    acc      = (float*)d_ws;
    unsigned* paircnt  = ((unsigned*)d_ws) + 8;
    float*    m        = ((float*)d_ws) + 16;

    zero_acc_kernel<<<1, 32, 0, stream>>>(acc);

    // 8 waves / block, one wave per 1400-class row
    disease_ce_kernel<<<(nRowsD + 7) / 8, 256, 0, stream>>>(
        disease_logits, disease_targets, acc, nRowsD);

    time_unc_kernel<<<(Nbs + 255) / 256, 256, 0, stream>>>(
        tte, ttg, uncertainty, acc, Nbs);

    risk_ce_kernel<<<(nRowsR + 255) / 256, 256, 0, stream>>>(
        risk_logits, risk_targets, acc, nRowsR);

    surv_mean_wmma_kernel<<<(n + 15) / 16, 32, 0, stream>>>(
        surv_curves, m, n, T);

    const int nT = (n + 255) / 256;
    dim3 grid2(nT, nT);
    concordance_kernel<<<grid2, 256, 0, stream>>>(
        m, surv_times, events, acc, paircnt, n);

    finalize_kernel<<<1, 32, 0, stream>>>(acc, paircnt, out, Nbs);
}